// CombinedLoss_29961691857676
// MI455X (gfx1250) — compile-verified
//
#include <hip/hip_runtime.h>
#include <hip/hip_bf16.h>
#include <stdint.h>

// CombinedLoss (weighted BCE via 9x9 morphology boundary weights + soft dice)
// B=32, C=1, H=W=1024, output = single f32 scalar.

#define NBATCH 32
#define IMH    1024
#define IMW    1024
#define TILE   64
#define HALO   4
#define PT     (TILE + 2*HALO)   // 72 (halo tile edge)
#define ST     76                // s_t row stride: mult of 4 (b128), 76%64=12 (banks ok)
#define RS     68                // row-filter stride: mult of 4, 68%64=4

typedef __attribute__((ext_vector_type(2))) float v2f;
typedef __attribute__((ext_vector_type(8))) float v8f;

#ifndef USE_ASYNC_LDS
#define USE_ASYNC_LDS 1          // gfx1250 async global->LDS path
#endif

#if defined(__has_builtin)
#if __has_builtin(__builtin_amdgcn_wmma_f32_16x16x4_f32)
#define HAVE_WMMA 1
#endif
#endif

__device__ __forceinline__ float max3f(float a, float b, float c) { return fmaxf(fmaxf(a, b), c); }
__device__ __forceinline__ float min3f(float a, float b, float c) { return fminf(fminf(a, b), c); }

__device__ __forceinline__ uint32_t lds_off(void* p) {
  // generic -> LDS(as3) addrspacecast; as3 pointer value == byte offset in LDS
  return (uint32_t)(uintptr_t)(__attribute__((address_space(3))) char*)(char*)p;
}

__global__ __launch_bounds__(256) void combined_loss_main(
    const float* __restrict__ logits, const float* __restrict__ targets,
    float* __restrict__ ws)
{
  __shared__ __align__(16) float s_t [PT * ST];   // halo tile of targets
  __shared__ __align__(16) float s_mx[PT * RS];   // horizontal 9-tap max
  __shared__ __align__(16) float s_mn[PT * RS];   // horizontal 9-tap min
  __shared__ float s_red[8][4];

  const int tid = threadIdx.x;
  const int b   = blockIdx.x >> 8;       // batch image
  const int tin = blockIdx.x & 255;      // tile in image (16x16 tiles)
  const int y0  = (tin >> 4) * TILE;
  const int x0  = (tin & 15) * TILE;

  const float* tgtb = targets + ((size_t)b << 20);
  const float* logb = logits  + ((size_t)b << 20);

  // ---- Phase 1: targets halo tile -> LDS (clamp == cv2 border-ignore) ----
  for (int i = tid; i < PT * PT; i += 256) {
    int r  = i / PT;
    int c  = i - r * PT;
    int gy = min(max(y0 - HALO + r, 0), IMH - 1);
    int gx = min(max(x0 - HALO + c, 0), IMW - 1);
#if USE_ASYNC_LDS
    uint32_t lo = lds_off(&s_t[r * ST + c]);
    uint32_t go = (uint32_t)((gy << 10) + gx) * 4u;
    asm volatile("global_load_async_to_lds_b32 %0, %1, %2"
                 :: "v"(lo), "v"(go), "s"(tgtb) : "memory");
#else
    s_t[r * ST + c] = tgtb[(gy << 10) + gx];
#endif
  }
#if USE_ASYNC_LDS
  asm volatile("s_wait_asynccnt 0" ::: "memory");
#endif
  __syncthreads();

  // ---- Phase 2: horizontal 9-tap max/min (register-blocked runs of 16) ----
  for (int w = tid; w < PT * 4; w += 256) {   // 72 rows x 4 strips of 16 cols
    int r  = w >> 2;
    int c0 = (w & 3) << 4;                    // 0,16,32,48 (16B aligned)
    float hv[24];
#pragma unroll
    for (int j4 = 0; j4 < 6; j4++) {          // 6x ds_load_b128
      float4 q = *(const float4*)&s_t[r * ST + c0 + j4 * 4];
      hv[j4*4+0] = q.x; hv[j4*4+1] = q.y; hv[j4*4+2] = q.z; hv[j4*4+3] = q.w;
    }
    float omx[16], omn[16];
#pragma unroll
    for (int k = 0; k < 16; k++) {
      omx[k] = max3f(max3f(hv[k],   hv[k+1], hv[k+2]),
                     max3f(hv[k+3], hv[k+4], hv[k+5]),
                     max3f(hv[k+6], hv[k+7], hv[k+8]));
      omn[k] = min3f(min3f(hv[k],   hv[k+1], hv[k+2]),
                     min3f(hv[k+3], hv[k+4], hv[k+5]),
                     min3f(hv[k+6], hv[k+7], hv[k+8]));
    }
#pragma unroll
    for (int q = 0; q < 4; q++) {             // 8x ds_store_b128
      *(float4*)&s_mx[r * RS + c0 + q*4] =
          make_float4(omx[q*4], omx[q*4+1], omx[q*4+2], omx[q*4+3]);
      *(float4*)&s_mn[r * RS + c0 + q*4] =
          make_float4(omn[q*4], omn[q*4+1], omn[q*4+2], omn[q*4+3]);
    }
  }
  __syncthreads();

  // ---- Phase 3: vertical 9-tap + BCE/sigmoid math + reductions ----
  // Lane = column (coalesced global loads of logits); thread owns 16 rows.
  const int c  = tid & 63;
  const int r0 = (tid >> 6) << 4;   // 0,16,32,48

  float vmx[24], vmn[24];
#pragma unroll
  for (int j = 0; j < 24; j++) {
    vmx[j] = s_mx[(r0 + j) * RS + c];
    vmn[j] = s_mn[(r0 + j) * RS + c];
  }

  const float LOG2E = 1.44269504088896340f;
  const float LN2   = 0.69314718055994531f;

  float accI = 0.f, accP = 0.f, accT = 0.f;
#if HAVE_WMMA
  v8f accC = {0.f, 0.f, 0.f, 0.f, 0.f, 0.f, 0.f, 0.f};
  const v2f onesB = {1.f, 1.f};
  float pend = 0.f;
  float accBCE;
#else
  float accBCE = 0.f;
#endif

#pragma unroll
  for (int k = 0; k < 16; k++) {
    float dil = max3f(max3f(vmx[k],   vmx[k+1], vmx[k+2]),
                      max3f(vmx[k+3], vmx[k+4], vmx[k+5]),
                      max3f(vmx[k+6], vmx[k+7], vmx[k+8]));
    float ero = min3f(min3f(vmn[k],   vmn[k+1], vmn[k+2]),
                      min3f(vmn[k+3], vmn[k+4], vmn[k+5]),
                      min3f(vmn[k+6], vmn[k+7], vmn[k+8]));
    float wgt = 1.f + 2.f * (dil - ero);    // 1 + (3-1)*boundary

    float t = s_t[(r0 + k + HALO) * ST + (c + HALO)];
    float x = __builtin_nontemporal_load(&logb[((y0 + r0 + k) << 10) + (x0 + c)]);

    // e = exp(-|x|) via raw v_exp_f32; args are range-safe (no libm fixups).
    float ax    = fabsf(x);
    float e     = __builtin_amdgcn_exp2f(-ax * LOG2E);
    float onepe = 1.f + e;                        // in (1, 2]
    float rc    = __builtin_amdgcn_rcpf(onepe);
    float l1p   = __builtin_amdgcn_logf(onepe) * LN2;   // log1p(e)
    float bce   = fmaxf(x, 0.f) - x * t + l1p;
    float p     = (x >= 0.f) ? rc : e * rc;       // sigmoid(x)

    accI += p * t;
    accP += p;
    accT += t;

    float wb = wgt * bce;
#if HAVE_WMMA
    if ((k & 1) == 0) {
      pend = wb;
    } else {
      v2f a; a.x = pend; a.y = wb;
      // D = A x ones(4x16) + C : every column of C accumulates the full sum.
      accC = __builtin_amdgcn_wmma_f32_16x16x4_f32(
          false, a, false, onesB, (short)0, accC, false, false);
    }
#else
    accBCE += wb;
#endif
  }

#if HAVE_WMMA
  accBCE = accC[0] + accC[1] + accC[2] + accC[3] +
           accC[4] + accC[5] + accC[6] + accC[7];
#endif

  // wave32 butterfly reductions
  for (int o = 16; o; o >>= 1) {
    accBCE += __shfl_xor(accBCE, o, 32);
    accI   += __shfl_xor(accI,   o, 32);
    accP   += __shfl_xor(accP,   o, 32);
    accT   += __shfl_xor(accT,   o, 32);
  }
#if HAVE_WMMA
  accBCE *= (1.f / 16.f);   // 16 identical columns in C (exact: /2^4)
#endif

  const int wv = tid >> 5;
  if ((tid & 31) == 0) {
    s_red[wv][0] = accBCE; s_red[wv][1] = accI;
    s_red[wv][2] = accP;   s_red[wv][3] = accT;
  }
  __syncthreads();
  if (tid < 4) {
    float v = 0.f;
#pragma unroll
    for (int wq = 0; wq < 8; wq++) v += s_red[wq][tid];
    // ws layout: [0]=bce_sum, [1+b]=I_b, [33+b]=P_b, [65+b]=T_b
    float* dst = (tid == 0) ? ws : (ws + 1 + (tid - 1) * NBATCH + b);
    atomicAdd(dst, v);
  }
}

__global__ void zero_ws_kernel(float* __restrict__ ws) {
  int i = threadIdx.x;
  if (i < 97) ws[i] = 0.f;
}

__global__ void combined_loss_finalize(const float* __restrict__ ws,
                                       float* __restrict__ out) {
  int b = threadIdx.x;  // 32 threads, one per batch image
  float I = ws[1 + b], P = ws[33 + b], T = ws[65 + b];
  float score = 2.f * (I + 1.f) / (P + T + 1.f);
  for (int o = 16; o; o >>= 1) score += __shfl_xor(score, o, 32);
  if (b == 0) {
    float bce = ws[0] * (1.f / 33554432.f);  // mean over 32*1024*1024
    out[0] = bce + (1.f - score * (1.f / 32.f));
  }
}

extern "C" void kernel_launch(void* const* d_in, const int* in_sizes, int n_in,
                              void* d_out, int out_size, void* d_ws, size_t ws_size,
                              hipStream_t stream) {
  const float* logits  = (const float*)d_in[0];
  const float* targets = (const float*)d_in[1];
  float* ws  = (float*)d_ws;
  float* out = (float*)d_out;
  (void)in_sizes; (void)n_in; (void)out_size; (void)ws_size;

  zero_ws_kernel<<<1, 128, 0, stream>>>(ws);
  combined_loss_main<<<NBATCH * 16 * 16, 256, 0, stream>>>(logits, targets, ws);
  combined_loss_finalize<<<1, 32, 0, stream>>>(ws, out);
}